// QuerySHLayerDeprecated_4114578670351
// MI455X (gfx1250) — compile-verified
//
#include <hip/hip_runtime.h>
#include <cstdint>

#define TILE 256

__device__ __forceinline__ uint32_t lds_addr32(const void* p) {
    // Generic pointer to __shared__: low 32 bits are the LDS byte offset
    // (HW adds the wave's LDS_BASE).
    return (uint32_t)(uintptr_t)p;
}

__device__ __forceinline__ void compute_edge(
    const float* __restrict__ pos,
    const float* __restrict__ query,
    const int*   __restrict__ esrc,
    const int*   __restrict__ edst,
    long long e, float sh[9], float em[10])
{
    // Index streams are read exactly once per call -> non-temporal loads.
    const int s = __builtin_nontemporal_load(&esrc[e]);
    const int d = __builtin_nontemporal_load(&edst[e]);

    // Gather tables (1.2 MB + 120 KB) are hot -> default (RT) caching.
    const float qx = query[3 * s + 0];
    const float qy = query[3 * s + 1];
    const float qz = query[3 * s + 2];
    const float x0 = qx - pos[3 * d + 0];
    const float y0 = qy - pos[3 * d + 1];
    const float z0 = qz - pos[3 * d + 2];

    const float r2 = x0 * x0 + y0 * y0 + z0 * z0;
    const float rn = (r2 > 0.0f) ? __builtin_amdgcn_rsqf(r2) : 0.0f;
    const float x = x0 * rn, y = y0 * rn, z = z0 * rn;
    const float L = r2 * rn;  // |vec| (0 when r2==0, matching reference)

    const float SQRT3 = 1.7320508075688772f;
    sh[0] = 1.0f;
    sh[1] = x;
    sh[2] = y;
    sh[3] = z;
    sh[4] = SQRT3 * x * z;
    sh[5] = SQRT3 * x * y;
    sh[6] = y * y - 0.5f * (x * x + z * z);
    sh[7] = SQRT3 * y * z;
    sh[8] = 0.8660254037844386f * (z * z - x * x);

    // smooth_finite_embed: step = 1/11, values v_i=(i+1)/11.
    // sus(a)*sus(b) = exp(-(1/a+1/b)) when a>0 && b>0, else 0.
    // SCALE = 1.14136 * e^2 * sqrt(10)
    const float SCALE = 26.669300f;
#pragma unroll
    for (int i = 0; i < 10; ++i) {
        const float vi = (float)(i + 1) * (1.0f / 11.0f);
        const float diff = (L - vi) * 11.0f;
        const float a = diff + 1.0f;
        const float b = 1.0f - diff;
        float f = 0.0f;
        if (a > 0.0f && b > 0.0f) {
            f = SCALE * __expf(-(__builtin_amdgcn_rcpf(a) +
                                 __builtin_amdgcn_rcpf(b)));
        }
        em[i] = f;
    }
}

__global__ __launch_bounds__(TILE) void edge_sh_embed_kernel(
    const float* __restrict__ pos,      // [100000,3]
    const float* __restrict__ query,    // [10000,3] (flattened 8x1250x3)
    const int*   __restrict__ esrc,     // [E]
    const int*   __restrict__ edst,     // [E]
    float* __restrict__ out_sh,         // [E,9]
    float* __restrict__ out_em,         // [E,10]
    int nEdges)
{
    __shared__ __align__(16) float lsh[TILE * 9];   // 9216 B, tile in global order
    __shared__ __align__(16) float lem[TILE * 10];  // 10240 B

    const int t = threadIdx.x;
    const long long base = (long long)blockIdx.x * TILE;
    const long long e = base + t;
    const bool fullTile = (base + TILE) <= (long long)nEdges;

    if (fullTile) {
        // EXEC is all-1s on this whole path (required for async LDS ops).
        float sh[9], em[10];
        compute_edge(pos, query, esrc, edst, e, sh, em);

        // Stage outputs in LDS in exact global memory order.
        // Strides 9 and 10 dwords are bank-conflict-free across a wave32.
#pragma unroll
        for (int k = 0; k < 9; ++k)  lsh[t * 9 + k]  = sh[k];
#pragma unroll
        for (int k = 0; k < 10; ++k) lem[t * 10 + k] = em[k];
        __syncthreads();

        // Stream LDS -> global as contiguous b128 chunks via the CDNA5 async
        // store path (no VGPR staging; tracked on ASYNCcnt). Output is
        // write-once -> non-temporal so the gather tables stay in L2.
        float* gsh = out_sh + base * 9;
        for (int c = t; c < (TILE * 9) / 4; c += TILE) {   // 576 chunks
            const uint64_t ga = (uint64_t)(uintptr_t)(gsh + c * 4);
            const uint32_t la = lds_addr32(&lsh[c * 4]);
            asm volatile("global_store_async_from_lds_b128 %0, %1, off th:TH_STORE_NT"
                         :: "v"(ga), "v"(la) : "memory");
        }
        float* gem = out_em + base * 10;
        for (int c = t; c < (TILE * 10) / 4; c += TILE) {  // 640 chunks
            const uint64_t ga = (uint64_t)(uintptr_t)(gem + c * 4);
            const uint32_t la = lds_addr32(&lem[c * 4]);
            asm volatile("global_store_async_from_lds_b128 %0, %1, off th:TH_STORE_NT"
                         :: "v"(ga), "v"(la) : "memory");
        }
        asm volatile("s_wait_asynccnt 0" ::: "memory");
    } else if (e < nEdges) {
        // Tail tile (only the last block, and only if E % TILE != 0):
        // direct scalar stores.
        float sh[9], em[10];
        compute_edge(pos, query, esrc, edst, e, sh, em);
#pragma unroll
        for (int k = 0; k < 9; ++k)  out_sh[e * 9 + k]  = sh[k];
#pragma unroll
        for (int k = 0; k < 10; ++k) out_em[e * 10 + k] = em[k];
    }
}

extern "C" void kernel_launch(void* const* d_in, const int* in_sizes, int n_in,
                              void* d_out, int out_size, void* d_ws, size_t ws_size,
                              hipStream_t stream) {
    const float* pos   = (const float*)d_in[0];
    const float* query = (const float*)d_in[1];
    const int*   esrc  = (const int*)d_in[2];
    const int*   edst  = (const int*)d_in[3];
    const int nEdges   = in_sizes[2];

    float* out_sh = (float*)d_out;
    float* out_em = out_sh + (size_t)nEdges * 9;

    const int blocks = (nEdges + TILE - 1) / TILE;
    edge_sh_embed_kernel<<<blocks, TILE, 0, stream>>>(
        pos, query, esrc, edst, out_sh, out_em, nEdges);
}